// BatchedMixedSphericalGaussianDistribution_21638045237433
// MI455X (gfx1250) — compile-verified
//
#include <hip/hip_runtime.h>

typedef float f32x4 __attribute__((ext_vector_type(4)));

#define KLOBES 8
#define PI_F 3.14159265358979323846f
#define INV_4PI_F 0.07957747154594767f
#define INV_2PI_F 0.15915494309189535f
#define LOG2E_F 1.4426950408889634f
#define LN2_F 0.6931471805599453f

// ---- single-instruction hardware transcendentals (with safe fallbacks) ----
#if __has_builtin(__builtin_amdgcn_rcpf)
__device__ __forceinline__ float frcp(float x) { return __builtin_amdgcn_rcpf(x); }
#else
__device__ __forceinline__ float frcp(float x) { return __fdividef(1.f, x); }
#endif

#if __has_builtin(__builtin_amdgcn_sqrtf)
__device__ __forceinline__ float fsqrt(float x) { return __builtin_amdgcn_sqrtf(x); }
#else
__device__ __forceinline__ float fsqrt(float x) { return sqrtf(x); }
#endif

#if __has_builtin(__builtin_amdgcn_exp2f)
__device__ __forceinline__ float fexp(float x) { return __builtin_amdgcn_exp2f(x * LOG2E_F); }
#else
__device__ __forceinline__ float fexp(float x) { return __expf(x); }
#endif

#if __has_builtin(__builtin_amdgcn_logf)
__device__ __forceinline__ float flog(float x) { return __builtin_amdgcn_logf(x) * LN2_F; }
#else
__device__ __forceinline__ float flog(float x) { return __logf(x); }
#endif

#if __has_builtin(__builtin_amdgcn_sinf) && __has_builtin(__builtin_amdgcn_cosf)
__device__ __forceinline__ void fsincos(float x, float* s, float* c) {
    const float r = x * INV_2PI_F;             // v_sin/v_cos take revolutions
    *s = __builtin_amdgcn_sinf(r);
    *c = __builtin_amdgcn_cosf(r);
}
#else
__device__ __forceinline__ void fsincos(float x, float* s, float* c) { __sincosf(x, s, c); }
#endif

__global__ __launch_bounds__(256) void bmsg_sample_pdf_kernel(
    const float* __restrict__ lam,
    const float* __restrict__ kappa,
    const float* __restrict__ theta,
    const float* __restrict__ phi,
    const float* __restrict__ u_cat,
    const float* __restrict__ u_w,
    const float* __restrict__ u_phi_s,
    float* __restrict__ out,
    int B)
{
    const int b = blockIdx.x * 256 + threadIdx.x;
    if (b >= B) return;
    const size_t row = (size_t)b * KLOBES;

    // ---- gfx1250 distance prefetch into L2 (global_prefetch_b8) ----
    const int PF = 8192; // rows ahead
    if (b + PF < B) {
        const size_t prow = row + (size_t)PF * KLOBES;
        __builtin_prefetch(lam   + prow, 0, 1);
        __builtin_prefetch(kappa + prow, 0, 1);
        __builtin_prefetch(theta + prow, 0, 1);
        __builtin_prefetch(phi   + prow, 0, 1);
        __builtin_prefetch(u_cat + prow, 0, 1);
    }

    // ---- issue all streaming loads up front (coalesced b128) ----
    f32x4 la0 = *(const f32x4*)(lam   + row);
    f32x4 la1 = *(const f32x4*)(lam   + row + 4);
    f32x4 ka0 = *(const f32x4*)(kappa + row);
    f32x4 ka1 = *(const f32x4*)(kappa + row + 4);
    f32x4 th0 = *(const f32x4*)(theta + row);
    f32x4 th1 = *(const f32x4*)(theta + row + 4);
    f32x4 ph0 = *(const f32x4*)(phi   + row);
    f32x4 ph1 = *(const f32x4*)(phi   + row + 4);
    f32x4 uc0 = *(const f32x4*)(u_cat + row);
    f32x4 uc1 = *(const f32x4*)(u_cat + row + 4);
    const float uw   = u_w[b];
    const float uphi = u_phi_s[b];

    float kav[KLOBES], thv[KLOBES], phv[KLOBES], ucv[KLOBES], weights[KLOBES];
    #pragma unroll
    for (int k = 0; k < 4; ++k) {
        kav[k] = ka0[k]; kav[k + 4] = ka1[k];
        thv[k] = th0[k]; thv[k + 4] = th1[k];
        phv[k] = ph0[k]; phv[k + 4] = ph1[k];
        ucv[k] = uc0[k]; ucv[k + 4] = uc1[k];
    }

    // ---- mixture weights (lambdas > 0 and total >= 1e-5 by construction,
    //      so the reference's isfinite/clip guards are identity and dropped) ----
    float total_pre = 0.f;
    float lamv[KLOBES];
    #pragma unroll
    for (int k = 0; k < 4; ++k) {
        lamv[k]     = fmaxf(la0[k], 0.f) + 1e-6f;
        lamv[k + 4] = fmaxf(la1[k], 0.f) + 1e-6f;
    }
    #pragma unroll
    for (int k = 0; k < KLOBES; ++k) total_pre += lamv[k];

    const float total     = fmaxf(total_pre, 1e-5f);
    const float inv_total = frcp(total);
    #pragma unroll
    for (int k = 0; k < KLOBES; ++k) weights[k] = lamv[k] * inv_total;

    // rs = sum_k weights[k] == total_pre / total  (in [0.8, 1]; degenerate
    // rs<=1e-12 branch of the reference is unreachable)
    const float rs     = total_pre * inv_total;
    const float inv_rs = frcp(rs + 1e-12f);

    // ---- Gumbel argmax over log(probs)+g (first-occurrence ties) ----
    int   idx  = 0;
    float best = 0.f;
    #pragma unroll
    for (int k = 0; k < KLOBES; ++k) {
        const float pk = weights[k] * inv_rs;
        const float g  = -flog(-flog(ucv[k] + 1e-12f) + 1e-12f);
        const float sc = flog(pk + 1e-20f) + g;
        if (k == 0 || sc > best) { best = sc; idx = k; }
    }

    // ---- select chosen lobe params ----
    float ck = kav[0], ct = thv[0], cp = phv[0];
    #pragma unroll
    for (int k = 1; k < KLOBES; ++k) {
        if (idx == k) { ck = kav[k]; ct = thv[k]; cp = phv[k]; }
    }

    // chosen mean direction
    float st_c, ctt, sp_c, cpp_c;
    fsincos(ct, &st_c, &ctt);
    fsincos(cp, &sp_c, &cpp_c);
    const float mux = st_c * cpp_c;
    const float muy = st_c * sp_c;
    const float muz = ctt;

    // ---- vMF polar angle ----
    const bool  kz     = (ck < 1e-5f);
    const float safe_k = kz ? 1.f : ck;
    const float w_uni  = 2.f * uw - 1.f;
    const float w_vmf  = 1.f + flog(uw + (1.f - uw) * fexp(-2.f * safe_k)) * frcp(safe_k);
    float wcos = kz ? w_uni : w_vmf;
    wcos = fminf(fmaxf(wcos, -1.f + 1e-5f), 1.f - 1e-5f);

    // ---- branchless orthonormal frame ----
    const float s  = (muz >= 0.f) ? 1.f : -1.f;
    const float a  = -frcp(s + muz);
    const float bb = mux * muy * a;
    const float v1x = 1.f + s * mux * mux * a;
    const float v1y = s * bb;
    const float v1z = -s * mux;
    const float v2x = bb;
    const float v2y = s + muy * muy * a;
    const float v2z = -muy;

    const float az = 2.f * PI_F * uphi;
    float saz, caz;
    fsincos(az, &saz, &caz);
    const float sin_th = fsqrt(1.f - wcos * wcos);  // wcos clipped -> arg >= ~2e-5

    const float sx = sin_th * (v1x * caz + v2x * saz) + wcos * mux;
    const float sy = sin_th * (v1y * caz + v2y * saz) + wcos * muy;
    const float sz = sin_th * (v1z * caz + v2z * saz) + wcos * muz;

    // ---- mixture PDF at the sample ----
    float pdf = 0.f;
    #pragma unroll
    for (int k = 0; k < KLOBES; ++k) {
        float stk, ctk, spk, cpk;
        fsincos(thv[k], &stk, &ctk);
        fsincos(phv[k], &spk, &cpk);
        const float mx = stk * cpk;
        const float my = stk * spk;
        const float mz = ctk;
        const float dot = mx * sx + my * sy + mz * sz;
        const bool  small = (kav[k] < 1e-5f);
        const float sk    = fmaxf(kav[k], 1e-6f);
        const float denom = 2.f * PI_F * (1.f - fexp(-2.f * sk));
        const float C     = small ? INV_4PI_F : sk * frcp(denom);
        pdf = fmaf(weights[k] * C, fexp(kav[k] * (dot - 1.f)), pdf);
    }

    // ---- non-temporal b128 store of [sample.xyz, pdf] ----
    f32x4 o;
    o[0] = sx; o[1] = sy; o[2] = sz; o[3] = pdf;
    __builtin_nontemporal_store(o, (f32x4*)(out + (size_t)b * 4));
}

extern "C" void kernel_launch(void* const* d_in, const int* in_sizes, int n_in,
                              void* d_out, int out_size, void* d_ws, size_t ws_size,
                              hipStream_t stream) {
    const float* lam     = (const float*)d_in[0];
    const float* kappa   = (const float*)d_in[1];
    const float* theta   = (const float*)d_in[2];
    const float* phi     = (const float*)d_in[3];
    const float* u_cat   = (const float*)d_in[4];
    const float* u_w     = (const float*)d_in[5];
    const float* u_phi_s = (const float*)d_in[6];
    float* out = (float*)d_out;

    const int B = in_sizes[5]; // u_w is (B,)
    dim3 block(256);
    dim3 grid((B + 255) / 256);
    bmsg_sample_pdf_kernel<<<grid, block, 0, stream>>>(
        lam, kappa, theta, phi, u_cat, u_w, u_phi_s, out, B);
}